// HNO_37065567764989
// MI455X (gfx1250) — compile-verified
//
#include <hip/hip_runtime.h>
#include <math.h>

#define MP1 524288
#define NXC 32768
#define PP  128
#define HTC 128
#define HEC 64

typedef float v2f __attribute__((ext_vector_type(2)));
typedef float v8f __attribute__((ext_vector_type(8)));

__device__ __forceinline__ float tanh_f(float x) {
#if defined(__HIP_DEVICE_COMPILE__)
# if __has_builtin(__builtin_amdgcn_tanhf)
  return __builtin_amdgcn_tanhf(x);
# elif __has_builtin(__builtin_amdgcn_tanh_f32)
  return __builtin_amdgcn_tanh_f32(x);
# else
  return tanhf(x);
# endif
#else
  return tanhf(x);
#endif
}

__device__ __forceinline__ v8f wmma4(v2f a, v2f b, v8f c) {
#if defined(__HIP_DEVICE_COMPILE__)
# if __has_builtin(__builtin_amdgcn_wmma_f32_16x16x4_f32)
  // 8 args: (neg_a, A, neg_b, B, c_mod, C, reuse_a, reuse_b)
  return __builtin_amdgcn_wmma_f32_16x16x4_f32(false, a, false, b, (short)0, c,
                                               false, false);
# else
#  error "missing __builtin_amdgcn_wmma_f32_16x16x4_f32 on this toolchain"
# endif
#else
  (void)a; (void)b;
  return c;
#endif
}

// ---------------------------------------------------------------------------
// Kernel 1: branch partial matvec.  grid = 128 rows * 8 chunks.
// Each block reduces 65536 elements of one row -> one partial in ws.
// ---------------------------------------------------------------------------
__global__ __launch_bounds__(256) void hno_branch_kernel(
    const float* __restrict__ Wb, const float* __restrict__ a,
    float* __restrict__ part) {
  __shared__ float red[256];
  const int p = blockIdx.x >> 3;
  const int r = blockIdx.x & 7;
  const float4* wb4 = (const float4*)(Wb + (size_t)p * MP1 + (size_t)r * 65536);
  const float4* a4  = (const float4*)(a + (size_t)r * 65536);
  float s = 0.f;
#pragma unroll 4
  for (int i = 0; i < 64; ++i) {
    float4 wv = wb4[i * 256 + threadIdx.x];
    float4 av = a4[i * 256 + threadIdx.x];
    s += wv.x * av.x + wv.y * av.y + wv.z * av.z + wv.w * av.w;
  }
  red[threadIdx.x] = s;
  __syncthreads();
  for (int st = 128; st > 0; st >>= 1) {
    if (threadIdx.x < st) red[threadIdx.x] += red[threadIdx.x + st];
    __syncthreads();
  }
  if (threadIdx.x == 0) part[p * 8 + r] = red[0];
}

// ---------------------------------------------------------------------------
// Kernel 2: b = sum(partials)+bb ; v = Wt3^T b ; c = b . bt3
// ---------------------------------------------------------------------------
__global__ __launch_bounds__(128) void hno_setup_kernel(
    const float* __restrict__ part, const float* __restrict__ bb,
    const float* __restrict__ Wt3, const float* __restrict__ bt3,
    float* __restrict__ vws, float* __restrict__ cws) {
  __shared__ float bL[128];
  __shared__ float cred[128];
  const int t = threadIdx.x;
  float s = bb[t];
  for (int r = 0; r < 8; ++r) s += part[t * 8 + r];
  bL[t] = s;
  __syncthreads();
  float vj = 0.f;
  for (int p = 0; p < PP; ++p) vj += bL[p] * Wt3[p * HTC + t];
  vws[t] = vj;
  cred[t] = bL[t] * bt3[t];
  __syncthreads();
  for (int st = 64; st > 0; st >>= 1) {
    if (t < st) cred[t] += cred[t + st];
    __syncthreads();
  }
  if (t == 0) cws[0] = cred[0];
}

// ---------------------------------------------------------------------------
// Kernel 3: fused trunk(3rd-order jets) + EnergyNet(3-jet in 2 vars) per point.
// Block = 256 threads = 8 wave32, 16 x-points per block.
// ---------------------------------------------------------------------------
__global__ __launch_bounds__(256) void hno_main_kernel(
    const float* __restrict__ x, const float* __restrict__ tptr,
    const float* __restrict__ Wt1, const float* __restrict__ bt1,
    const float* __restrict__ Wt2, const float* __restrict__ bt2,
    const float* __restrict__ We1, const float* __restrict__ be1,
    const float* __restrict__ We2, const float* __restrict__ be2,
    const float* __restrict__ We3, const float* __restrict__ vws,
    const float* __restrict__ cws, float* __restrict__ out) {
  // 80 KB aliased buffer: phase A = trunk H1 jets [4][128*16] (32 KB);
  // phase B = energy G1 jets [64][160] (40 KB) overlaying dead H1, plus
  // S2 jets [64][160] (40 KB) at offset 10240.
  __shared__ float sbuf[20480];
  __shared__ float red1[4][16][16];
  __shared__ float red2[10][16][16];
  __shared__ float usS[4][16];
  __shared__ float xsS[16];
  __shared__ float vvL[128];
  __shared__ float bt2L[128];
  __shared__ float be2L[64];
  __shared__ float we3L[64];
  __shared__ float FcS[10][16];
  __shared__ float ccL[1];

  const int tid  = threadIdx.x;
  const int lane = tid & 31;
  const int w    = tid >> 5;   // wave id 0..7
  const int n16  = lane & 15;  // point / M-row within tile
  const int g    = lane >> 4;  // lane-group (selects K pair / M half)

  if (tid < 16) xsS[tid] = x[blockIdx.x * 16 + tid];
  if (tid < 128) {
    vvL[tid]  = vws[tid];
    bt2L[tid] = bt2[tid];
  }
  if (tid < 64) {
    be2L[tid] = be2[tid];
    we3L[tid] = We3[tid];
  }
  if (tid == 0) ccL[0] = cws[0];
  const float tval = tptr[0];
  __syncthreads();

  // ---- trunk layer 1: analytic jets (z' = w0 const) -> Hs[4][128*16] ----
  {
    float* Hs = sbuf;  // 4 planes of 2048
#pragma unroll
    for (int i = 0; i < 8; ++i) {
      int item = tid + i * 256;  // 2048 = 128 neurons * 16 points
      int k = item >> 4, n = item & 15;
      float w0 = Wt1[2 * k], w1 = Wt1[2 * k + 1];
      float z  = w0 * xsS[n] + (w1 * tval + bt1[k]);
      float h  = tanh_f(z);
      float T1 = 1.f - h * h;
      float T2 = -2.f * h * T1;
      float T3 = -2.f * T1 * (1.f - 3.f * h * h);
      int o = k * 16 + n;
      Hs[0 * 2048 + o] = h;
      Hs[1 * 2048 + o] = T1 * w0;
      Hs[2 * 2048 + o] = T2 * w0 * w0;
      Hs[3 * 2048 + o] = T3 * w0 * w0 * w0;
    }
  }
  __syncthreads();

  // ---- trunk layer 2 via WMMA: wave w owns rows 16w..16w+15 ----
  // A(m,k): lane=(k/2)*16+m, vgpr=k%2 ; B(k,n): lane=(k/2)*16+n, vgpr=k%2
  float2 Areg[32];
  {
    const int rowA = 16 * w + n16;
#pragma unroll
    for (int kk = 0; kk < 32; ++kk) {
      int kb = 4 * kk + 2 * g;
      Areg[kk].x = Wt2[rowA * HTC + kb];
      Areg[kk].y = Wt2[rowA * HTC + kb + 1];
    }
  }
  v8f acc[4];
  for (int s = 0; s < 4; ++s) {
    const float* Hs = sbuf + s * 2048;
    v8f c = {0.f, 0.f, 0.f, 0.f, 0.f, 0.f, 0.f, 0.f};
#pragma unroll
    for (int kk = 0; kk < 32; ++kk) {
      int kb = 4 * kk + 2 * g;
      v2f a; a.x = Areg[kk].x; a.y = Areg[kk].y;
      v2f b; b.x = Hs[kb * 16 + n16]; b.y = Hs[(kb + 1) * 16 + n16];
      c = wmma4(a, b, c);
    }
    acc[s] = c;
  }

  // ---- tanh jet composition + partial dot with v ----
  {
    float part[4] = {0.f, 0.f, 0.f, 0.f};
#pragma unroll
    for (int j = 0; j < 8; ++j) {
      int row = 16 * w + j + 8 * g;  // C layout: vgpr j = M=j (lanes<16) / j+8
      float g0 = acc[0][j] + bt2L[row];
      float g1 = acc[1][j], g2v = acc[2][j], g3 = acc[3][j];
      float h  = tanh_f(g0);
      float T1 = 1.f - h * h;
      float T2 = -2.f * h * T1;
      float T3 = -2.f * T1 * (1.f - 3.f * h * h);
      float h0 = h;
      float h1 = T1 * g1;
      float h2 = T1 * g2v + T2 * g1 * g1;
      float h3 = T1 * g3 + 3.f * T2 * g1 * g2v + T3 * g1 * g1 * g1;
      float vr = vvL[row];
      part[0] += vr * h0; part[1] += vr * h1;
      part[2] += vr * h2; part[3] += vr * h3;
    }
    int slot = w * 2 + g;
#pragma unroll
    for (int s = 0; s < 4; ++s) red1[s][n16][slot] = part[s];
  }
  __syncthreads();

  if (tid < 64) {  // deterministic reduce -> u, u_x, u_xx, u_xxx
    int s = tid >> 4, n = tid & 15;
    float sum = (s == 0) ? ccL[0] : 0.f;
    for (int q = 0; q < 16; ++q) sum += red1[s][n][q];
    usS[s][n] = sum;
  }
  __syncthreads();

  // ---- energy layer 1: 3-jet in (y,z), trivial input jets -> G1[64][160] ---
  {
    float* G1 = sbuf;  // overlays dead trunk H1
#pragma unroll
    for (int i = 0; i < 4; ++i) {
      int item = tid + i * 256;  // 1024 = 64 neurons * 16 points
      int k = item >> 4, n = item & 15;
      float y = usS[0][n], z = usS[1][n];
      float av = We1[2 * k], bv = We1[2 * k + 1];
      float sv = av * y + bv * z + be1[k];
      float h  = tanh_f(sv);
      float T1 = 1.f - h * h;
      float T2 = -2.f * h * T1;
      float T3 = -2.f * T1 * (1.f - 3.f * h * h);
      float* gp = &G1[k * 160 + n];
      gp[0 * 16] = h;
      gp[1 * 16] = T1 * av;
      gp[2 * 16] = T1 * bv;
      gp[3 * 16] = T2 * av * av;
      gp[4 * 16] = T2 * av * bv;
      gp[5 * 16] = T2 * bv * bv;
      gp[6 * 16] = T3 * av * av * av;
      gp[7 * 16] = T3 * av * av * bv;
      gp[8 * 16] = T3 * av * bv * bv;
      gp[9 * 16] = T3 * bv * bv * bv;
    }
  }
  __syncthreads();

  // ---- energy layer 2 via WMMA: 4 M-tiles x 10 jet-tiles = 40, 5/wave ----
  {
    const float* G1 = sbuf;
    float* S2 = sbuf + 10240;
    for (int i = 0; i < 5; ++i) {
      int tt = w * 5 + i;
      int m = tt / 10, nt = tt % 10;
      int row = 16 * m + n16;
      v8f c = {0.f, 0.f, 0.f, 0.f, 0.f, 0.f, 0.f, 0.f};
#pragma unroll
      for (int kk = 0; kk < 16; ++kk) {
        int kb = 4 * kk + 2 * g;
        v2f a; a.x = We2[row * HEC + kb]; a.y = We2[row * HEC + kb + 1];
        v2f b; b.x = G1[kb * 160 + nt * 16 + n16];
        b.y = G1[(kb + 1) * 160 + nt * 16 + n16];
        c = wmma4(a, b, c);
      }
#pragma unroll
      for (int j = 0; j < 8; ++j) {
        int r = 16 * m + j + 8 * g;
        S2[r * 160 + nt * 16 + n16] = c[j];
      }
    }
  }
  __syncthreads();

  // ---- tanh 3-jet composition + layer-3 dot (deterministic tree) ----
  {
    const float* S2 = sbuf + 10240;
    int n = tid & 15, grp = tid >> 4;
    float pc[10];
#pragma unroll
    for (int c = 0; c < 10; ++c) pc[c] = 0.f;
#pragma unroll
    for (int i = 0; i < 4; ++i) {
      int k = grp * 4 + i;
      const float* sp = &S2[k * 160 + n];
      float g0 = sp[0] + be2L[k];
      float gy = sp[16], gz = sp[32];
      float gyy = sp[48], gyz = sp[64], gzz = sp[80];
      float gyyy = sp[96], gyyz = sp[112], gyzz = sp[128], gzzz = sp[144];
      float v  = tanh_f(g0);
      float T1 = 1.f - v * v;
      float T2 = -2.f * v * T1;
      float T3 = -2.f * T1 * (1.f - 3.f * v * v);
      float h[10];
      h[0] = v;
      h[1] = T1 * gy;
      h[2] = T1 * gz;
      h[3] = T1 * gyy + T2 * gy * gy;
      h[4] = T1 * gyz + T2 * gy * gz;
      h[5] = T1 * gzz + T2 * gz * gz;
      h[6] = T1 * gyyy + 3.f * T2 * gy * gyy + T3 * gy * gy * gy;
      h[7] = T1 * gyyz + T2 * (2.f * gy * gyz + gz * gyy) + T3 * gy * gy * gz;
      h[8] = T1 * gyzz + T2 * (2.f * gz * gyz + gy * gzz) + T3 * gy * gz * gz;
      h[9] = T1 * gzzz + 3.f * T2 * gz * gzz + T3 * gz * gz * gz;
      float w3 = we3L[k];
#pragma unroll
      for (int c = 0; c < 10; ++c) pc[c] += w3 * h[c];
    }
#pragma unroll
    for (int c = 0; c < 10; ++c) red2[c][n][grp] = pc[c];
  }
  __syncthreads();

  if (tid < 160) {
    int c = tid >> 4, n = tid & 15;
    float s = 0.f;
    for (int q = 0; q < 16; ++q) s += red2[c][n][q];
    FcS[c][n] = s;
  }
  __syncthreads();

  if (tid < 16) {
    int n = tid;
    float ux = usS[1][n], uxx = usS[2][n], uxxx = usS[3][n];
    float Fyy = FcS[3][n], Fyz = FcS[4][n], Fzz = FcS[5][n];
    float Fyyz = FcS[7][n], Fyzz = FcS[8][n], Fzzz = FcS[9][n];
    float Fyx  = Fyy * ux + Fyz * uxx;
    float Fzxx = Fyz * uxx + Fzz * uxxx + Fyyz * ux * ux +
                 2.f * Fyzz * ux * uxx + Fzzz * uxx * uxx;
    out[blockIdx.x * 16 + n] = -Fyx + Fzxx;
  }
}

// ---------------------------------------------------------------------------
extern "C" void kernel_launch(void* const* d_in, const int* in_sizes, int n_in,
                              void* d_out, int out_size, void* d_ws,
                              size_t ws_size, hipStream_t stream) {
  (void)in_sizes; (void)n_in; (void)ws_size;
  const float* a   = (const float*)d_in[0];
  const float* x   = (const float*)d_in[1];
  const float* t   = (const float*)d_in[2];
  const float* Wb  = (const float*)d_in[3];
  const float* bb  = (const float*)d_in[4];
  const float* Wt1 = (const float*)d_in[5];
  const float* bt1 = (const float*)d_in[6];
  const float* Wt2 = (const float*)d_in[7];
  const float* bt2 = (const float*)d_in[8];
  const float* Wt3 = (const float*)d_in[9];
  const float* bt3 = (const float*)d_in[10];
  const float* We1 = (const float*)d_in[11];
  const float* be1 = (const float*)d_in[12];
  const float* We2 = (const float*)d_in[13];
  const float* be2 = (const float*)d_in[14];
  const float* We3 = (const float*)d_in[15];
  float* out = (float*)d_out;

  float* wsf  = (float*)d_ws;
  float* part = wsf;          // 1024 floats
  float* vws  = wsf + 1024;   // 128 floats
  float* cws  = wsf + 1152;   // 1 float

  hno_branch_kernel<<<PP * 8, 256, 0, stream>>>(Wb, a, part);
  hno_setup_kernel<<<1, 128, 0, stream>>>(part, bb, Wt3, bt3, vws, cws);
  hno_main_kernel<<<out_size / 16, 256, 0, stream>>>(
      x, t, Wt1, bt1, Wt2, bt2, We1, be1, We2, be2, We3, vws, cws, out);
}